// ST_BIFNeuron_MS_81286551044604
// MI455X (gfx1250) — compile-verified
//
#include <hip/hip_runtime.h>

// ST-BIF multi-step spiking neuron, T=8, T_MAX=8, T_MIN=-8.
// Pure streaming elementwise op (~51 MB traffic) -> HBM-bound on MI455X.
// Strategy:
//   * float4 traffic; gfx1250 async global->LDS staging
//     (global_load_async_to_lds_b128 / ASYNCcnt / s_wait_asynccnt),
//     depth-3 software pipeline over 6 LDS buffers.
//   * recurrence fully unrolled in VGPRs; float Tt timer replaced by two
//     boolean lane-masks (bit-exact for T=8, see analysis) to cut VALU.
//   * fast path uses one base address + 24-bit instruction offsets.

#define TPB   256   // 8 wave32 per block
#define ITERS 8     // float4 chunks per thread
#define PF    3     // prefetch depth (async loads in flight per wave)
#define NBUF  6     // LDS buffers (reuse distance = NBUF - PF = 3 iters)
#define CHUNK_STRIDE_B (TPB * 16)  // 4096 B between a thread's chunks

// ---- gfx1250 async-tensor path helpers -------------------------------------

__device__ __forceinline__ void async_ld16(unsigned ldsAddr,
                                           unsigned long long gaddr) {
  asm volatile("global_load_async_to_lds_b128 %0, %1, off"
               :: "v"(ldsAddr), "v"(gaddr)
               : "memory");
}

template <int OFF>
__device__ __forceinline__ void async_ld16_off(unsigned ldsAddr,
                                               unsigned long long gaddr) {
  asm volatile("global_load_async_to_lds_b128 %0, %1, off offset:%2"
               :: "v"(ldsAddr), "v"(gaddr), "n"(OFF)
               : "memory");
}

// Dispatch an unroll-constant chunk index to a constant instruction offset.
__device__ __forceinline__ void issue_fast(int k, unsigned ldsAddr,
                                           unsigned long long gaddr) {
  switch (k) {
    case 0: async_ld16_off<0 * CHUNK_STRIDE_B>(ldsAddr, gaddr); break;
    case 1: async_ld16_off<1 * CHUNK_STRIDE_B>(ldsAddr, gaddr); break;
    case 2: async_ld16_off<2 * CHUNK_STRIDE_B>(ldsAddr, gaddr); break;
    case 3: async_ld16_off<3 * CHUNK_STRIDE_B>(ldsAddr, gaddr); break;
    case 4: async_ld16_off<4 * CHUNK_STRIDE_B>(ldsAddr, gaddr); break;
    case 5: async_ld16_off<5 * CHUNK_STRIDE_B>(ldsAddr, gaddr); break;
    case 6: async_ld16_off<6 * CHUNK_STRIDE_B>(ldsAddr, gaddr); break;
    default: async_ld16_off<7 * CHUNK_STRIDE_B>(ldsAddr, gaddr); break;
  }
}

template <int N>
__device__ __forceinline__ void wait_asynccnt() {
  asm volatile("s_wait_asynccnt %0" :: "n"(N) : "memory");
}

// remaining = min(PF, ITERS-1-k); k is an unroll constant so this folds.
__device__ __forceinline__ void constexpr_wait_impl(int remaining) {
  switch (remaining) {
    case 0:  wait_asynccnt<0>(); break;
    case 1:  wait_asynccnt<1>(); break;
    case 2:  wait_asynccnt<2>(); break;
    default: wait_asynccnt<3>(); break;
  }
}
#define constexpr_wait(k)                                                \
  constexpr_wait_impl(((ITERS - 1 - (k)) < PF) ? (ITERS - 1 - (k)) : PF)

// ---- the neuron recurrence (all in registers) ------------------------------
//
// Bit-exact reformulation for T=8:
//   (Tt < T_MAX)  <=>  no negative spike emitted so far       (!negSeen)
//   (Tt > T_MIN)  <=>  previous step did not emit a +1 spike  (!lastPos)
// All Tt values in the window are small integers, so the logical
// equivalence holds in float arithmetic too. v-update uses a single FMA:
// s*q is exact for s in {-1,0,+1}, so fma(-s,q,H) == H - s*q bit-exactly.

__device__ __forceinline__ float st_bif_elem(float x, float q) {
  float v = 0.0f, acc = 0.0f;
  bool negSeen = false, lastPos = false;
#pragma unroll
  for (int t = 0; t < 8; ++t) {
    float H      = v + x;
    bool  pos    = (H >= q) && !negSeen;
    bool  negRaw = (H < 0.0f) && !lastPos;
    float s      = pos ? 1.0f : (negRaw ? -1.0f : 0.0f);
    v = __builtin_fmaf(-s, q, H);
    acc += s;
    negSeen = negSeen || (negRaw && !pos);
    lastPos = pos;
  }
  return acc * 0.125f; // mean over T=8
}

// ---- per-block pipelines ----------------------------------------------------

__device__ __forceinline__ float4 neuron4(float4 xv, float q) {
  float4 r;
  r.x = st_bif_elem(xv.x, q);
  r.y = st_bif_elem(xv.y, q);
  r.z = st_bif_elem(xv.z, q);
  r.w = st_bif_elem(xv.w, q);
  return r;
}

// ---- kernel ----------------------------------------------------------------

__global__ void __launch_bounds__(TPB)
st_bif_ms_kernel(const float* __restrict__ x, const float* __restrict__ qp,
                 float* __restrict__ out, long long n4, long long n) {
  __shared__ float4 stage[NBUF][TPB];

  const float q = qp[0];
  const float4* __restrict__ x4 = reinterpret_cast<const float4*>(x);
  float4* __restrict__       o4 = reinterpret_cast<float4*>(out);

  if (n4 > 0) {  // uniform across the whole grid
    const long long blockBase = (long long)blockIdx.x * (long long)(TPB * ITERS);
    const long long base      = blockBase + threadIdx.x;

    // LDS byte address of this thread's slot in buffer 0 (low 32 bits of the
    // flat address of a __shared__ object are the LDS byte offset).
    const unsigned lds0 = (unsigned)(size_t)(&stage[0][threadIdx.x]);

    const bool fullBlock = (blockBase + (long long)(TPB * ITERS)) <= n4;  // block-uniform

    if (fullBlock) {
      // ---- fast path: single base address, immediate offsets ----
      const unsigned long long g0 = (unsigned long long)(size_t)(x4 + base);
#pragma unroll
      for (int k = 0; k < PF; ++k)
        issue_fast(k, lds0 + (unsigned)((k % NBUF) * CHUNK_STRIDE_B), g0);
#pragma unroll
      for (int k = 0; k < ITERS; ++k) {
        if (k + PF < ITERS)
          issue_fast(k + PF,
                     lds0 + (unsigned)(((k + PF) % NBUF) * CHUNK_STRIDE_B), g0);
        constexpr_wait(k);
        float4 xv = stage[k % NBUF][threadIdx.x];  // ds_load_b128
        o4[base + (long long)k * TPB] = neuron4(xv, q);  // global_store_b128
      }
    } else {
      // ---- boundary path: clamped addresses, predicated stores ----
      auto chunk_idx = [&](int k) -> long long {
        long long i = base + (long long)k * TPB;
        return (i < n4) ? i : (n4 - 1);
      };
#pragma unroll
      for (int k = 0; k < PF; ++k)
        async_ld16(lds0 + (unsigned)((k % NBUF) * CHUNK_STRIDE_B),
                   (unsigned long long)(size_t)(x4 + chunk_idx(k)));
#pragma unroll
      for (int k = 0; k < ITERS; ++k) {
        if (k + PF < ITERS)
          async_ld16(lds0 + (unsigned)(((k + PF) % NBUF) * CHUNK_STRIDE_B),
                     (unsigned long long)(size_t)(x4 + chunk_idx(k + PF)));
        constexpr_wait(k);
        float4 xv = stage[k % NBUF][threadIdx.x];
        long long i = base + (long long)k * TPB;
        if (i < n4) o4[i] = neuron4(xv, q);
      }
    }
  }

  // Scalar tail for n % 4 elements (0 for the reference shape).
  long long tailStart = n4 * 4;
  if (blockIdx.x == 0 && (long long)threadIdx.x < (n - tailStart)) {
    out[tailStart + threadIdx.x] = st_bif_elem(x[tailStart + threadIdx.x], q);
  }
}

// ---- launch ----------------------------------------------------------------

extern "C" void kernel_launch(void* const* d_in, const int* in_sizes, int n_in,
                              void* d_out, int out_size, void* d_ws,
                              size_t ws_size, hipStream_t stream) {
  const float* x  = (const float*)d_in[0];
  const float* q  = (const float*)d_in[1];
  float*       o  = (float*)d_out;

  long long n  = (long long)in_sizes[0];
  long long n4 = n / 4;

  long long chunksPerBlock = (long long)TPB * ITERS;
  long long blocks = (n4 + chunksPerBlock - 1) / chunksPerBlock;
  if (blocks < 1) blocks = 1;

  st_bif_ms_kernel<<<(int)blocks, TPB, 0, stream>>>(x, q, o, n4, n);
}